// InfoNCE2_22806276342427
// MI455X (gfx1250) — compile-verified
//
#include <hip/hip_runtime.h>
#include <hip/hip_bf16.h>
#include <math.h>

// Problem constants (reference: N=1024, X_DIM=64, Y_DIM=64, HID=64)
#define NN    1024
#define DIMX  64
#define HID   64
#define DIN   128   // X_DIM + Y_DIM

typedef __attribute__((ext_vector_type(2))) float v2f;
typedef __attribute__((ext_vector_type(8))) float v8f;

// ---------------------------------------------------------------------------
// Kernel 1: px = x @ W1x^T (+b1 folded), py = y @ W1y^T via V_WMMA_F32_16X16X4_F32.
// One wave32 per 16x16 output tile; K=64 swept in 16 chunks of 4.
// A layout (16x4 f32): lane l holds row M=l&15; reg r holds K = r + 2*(l>=16) (+k0).
// B layout (4x16 f32): lane l holds col N=l&15; reg r holds K = r + 2*(l>=16) (+k0).
// C/D layout (16x16 f32): reg r, lane l -> M = r + 8*(l>=16), N = l&15.
// ---------------------------------------------------------------------------
__global__ __launch_bounds__(32)
void infonce_gemm_wmma(const float* __restrict__ x, const float* __restrict__ y,
                       const float* __restrict__ W1, const float* __restrict__ b1,
                       float* __restrict__ pxb, float* __restrict__ pyo)
{
    const int lane = threadIdx.x;        // 0..31 (one full wave32; EXEC all ones)
    const int nb   = blockIdx.x >> 2;    // 0..63  : n-block
    const int hb   = blockIdx.x & 3;     // 0..3   : h-block
    const int n0   = nb * 16;
    const int h0   = hb * 16;
    const int half = lane >> 4;          // 0 or 1
    const int lo   = lane & 15;
    const int arow = n0 + lo;            // A-matrix row owned by this lane
    const int brow = h0 + lo;            // B-matrix column (= W1 row) owned by this lane

    v8f cx = {};
    v8f cy = {};

    #pragma unroll
    for (int kc = 0; kc < 16; ++kc) {
        const int kb = kc * 4 + half * 2;    // K offset for this lane's half

        v2f ax; ax[0] = x[arow * DIMX + kb];            ax[1] = x[arow * DIMX + kb + 1];
        v2f bx; bx[0] = W1[brow * DIN  + kb];           bx[1] = W1[brow * DIN  + kb + 1];
        cx = __builtin_amdgcn_wmma_f32_16x16x4_f32(
                 false, ax, false, bx, (short)0, cx, false, false);

        v2f ay; ay[0] = y[arow * DIMX + kb];            ay[1] = y[arow * DIMX + kb + 1];
        v2f by; by[0] = W1[brow * DIN + DIMX + kb];     by[1] = W1[brow * DIN + DIMX + kb + 1];
        cy = __builtin_amdgcn_wmma_f32_16x16x4_f32(
                 false, ay, false, by, (short)0, cy, false, false);
    }

    const float bias = b1[h0 + lo];
    #pragma unroll
    for (int r = 0; r < 8; ++r) {
        const int n = n0 + r + half * 8;
        const int h = h0 + lo;
        pxb[n * HID + h] = cx[r] + bias;   // px + b1 (bias folded once here)
        pyo[n * HID + h] = cy[r];
    }
}

__device__ __forceinline__ float softplus_f(float t) {
    // log1p(exp(t)) stable: max(t,0) + log1p(exp(-|t|))
    return fmaxf(t, 0.0f) + log1pf(expf(-fabsf(t)));
}

// ---------------------------------------------------------------------------
// Kernel 2: pairwise T1 with fused logsumexp partials.
// Grid: 256 blocks x 256 threads. Block b: i-block (b>>2) (16 rows), j-split (b&3)
// (256 columns, one j per thread). Each thread keeps pxb[j,:] in 64 VGPRs and
// reuses it across all 16 i's. Per row: wave32 shfl_xor (m,s) logsumexp reduce
// (no barriers), then one barrier + serial 8-wave merge -> partial (m,s).
// Diagonal T1[n,n] == T0[n] -> diag_out.
// ---------------------------------------------------------------------------
__global__ __launch_bounds__(256)
void infonce_pairwise(const float* __restrict__ pxb, const float* __restrict__ pyw,
                      const float* __restrict__ W2, const float* __restrict__ b2,
                      float* __restrict__ pm, float* __restrict__ ps,
                      float* __restrict__ diag_out)
{
    __shared__ float py_s[16 * HID];   // 4 KB: this block's 16 py rows
    __shared__ float w2_s[HID];
    __shared__ float red_m[16 * 8];    // [row][wave]
    __shared__ float red_s[16 * 8];

    const int tid   = threadIdx.x;
    const int i0    = (blockIdx.x >> 2) * 16;
    const int split = blockIdx.x & 3;
    const int j     = split * 256 + tid;     // this thread's column, 0..1023

    #pragma unroll
    for (int k = 0; k < 4; ++k) {
        const int idx = tid + k * 256;       // 0..1023
        py_s[idx] = pyw[i0 * HID + idx];
    }
    if (tid < HID) w2_s[tid] = W2[tid];
    const float b2v = b2[0];
    __syncthreads();

    // pxb row for this thread's j, held in registers for all 16 i's.
    float a[HID];
    #pragma unroll
    for (int h = 0; h < HID; h += 4) {
        const float4 v = *(const float4*)(pxb + j * HID + h);
        a[h] = v.x; a[h + 1] = v.y; a[h + 2] = v.z; a[h + 3] = v.w;
    }

    const int wave = tid >> 5;
    const int lane = tid & 31;

    #pragma unroll
    for (int ii = 0; ii < 16; ++ii) {
        float t = b2v;
        #pragma unroll
        for (int h = 0; h < HID; ++h)
            t = fmaf(fmaxf(a[h] + py_s[ii * HID + h], 0.0f), w2_s[h], t);

        const float v = softplus_f(t);
        if (i0 + ii == j) diag_out[j] = v;   // T0 diagonal element

        // Intra-wave logsumexp reduction; each lane starts with (m=v, s=1).
        float mm = v, ss = 1.0f;
        #pragma unroll
        for (int off = 16; off > 0; off >>= 1) {
            const float mo = __shfl_xor(mm, off);
            const float so = __shfl_xor(ss, off);
            const float mn = fmaxf(mm, mo);
            ss = ss * expf(mm - mn) + so * expf(mo - mn);
            mm = mn;
        }
        if (lane == 0) { red_m[ii * 8 + wave] = mm; red_s[ii * 8 + wave] = ss; }
    }
    __syncthreads();

    // Merge the 8 wave partials per row (threads 0..15), fixed order.
    if (tid < 16) {
        const int ii = tid;
        float mm = red_m[ii * 8], ss = red_s[ii * 8];
        #pragma unroll
        for (int w = 1; w < 8; ++w) {
            const float mo = red_m[ii * 8 + w], so = red_s[ii * 8 + w];
            const float mn = fmaxf(mm, mo);
            ss = ss * expf(mm - mn) + so * expf(mo - mn);
            mm = mn;
        }
        pm[split * NN + i0 + ii] = mm;
        ps[split * NN + i0 + ii] = ss;
    }
}

// ---------------------------------------------------------------------------
// Kernel 3: merge j-split partials -> lse[i]; deterministic final reduction.
// ---------------------------------------------------------------------------
__global__ __launch_bounds__(256)
void infonce_finalize(const float* __restrict__ pm, const float* __restrict__ ps,
                      const float* __restrict__ diag, float* __restrict__ out)
{
    __shared__ float r1[256];
    __shared__ float r2[256];
    const int tid = threadIdx.x;
    float sl = 0.0f, sd = 0.0f;
    for (int i = tid; i < NN; i += 256) {
        float mm = pm[i], ss = ps[i];
        #pragma unroll
        for (int sp = 1; sp < 4; ++sp) {
            const float mo = pm[sp * NN + i], so = ps[sp * NN + i];
            const float mn = fmaxf(mm, mo);
            ss = ss * expf(mm - mn) + so * expf(mo - mn);
            mm = mn;
        }
        sl += mm + logf(ss);     // lse[i]
        sd += diag[i];
    }
    r1[tid] = sl; r2[tid] = sd;
    __syncthreads();
    for (int off = 128; off > 0; off >>= 1) {
        if (tid < off) { r1[tid] += r1[tid + off]; r2[tid] += r2[tid + off]; }
        __syncthreads();
    }
    if (tid == 0)
        out[0] = r2[0] / (float)NN - (r1[0] / (float)NN - logf((float)NN));
}

extern "C" void kernel_launch(void* const* d_in, const int* in_sizes, int n_in,
                              void* d_out, int out_size, void* d_ws, size_t ws_size,
                              hipStream_t stream) {
    (void)in_sizes; (void)n_in; (void)out_size; (void)ws_size;

    const float* x  = (const float*)d_in[0];  // (1024,64)
    const float* y  = (const float*)d_in[1];  // (1024,64)
    /* d_in[2] = mask: all-ones, unused by the reference computation */
    const float* W1 = (const float*)d_in[3];  // (64,128) row-major
    const float* b1 = (const float*)d_in[4];  // (64,)
    const float* W2 = (const float*)d_in[5];  // (1,64)
    const float* b2 = (const float*)d_in[6];  // (1,)
    float* out = (float*)d_out;

    char* ws = (char*)d_ws;
    float* pxb  = (float*)(ws);                               // 1024*64 f32
    float* pyw  = (float*)(ws + (size_t)NN * HID * 4);        // 1024*64 f32
    float* pm   = (float*)(ws + (size_t)2 * NN * HID * 4);    // 4*1024 f32
    float* psum = (float*)(ws + (size_t)2 * NN * HID * 4 + (size_t)4 * NN * 4); // 4*1024 f32
    float* diag = (float*)(ws + (size_t)2 * NN * HID * 4 + (size_t)8 * NN * 4); // 1024 f32

    infonce_gemm_wmma<<<256, 32, 0, stream>>>(x, y, W1, b1, pxb, pyw);
    infonce_pairwise <<<256, 256, 0, stream>>>(pxb, pyw, W2, b2, pm, psum, diag);
    infonce_finalize <<<1, 256, 0, stream>>>(pm, psum, diag, out);
}